// LaplacianGeneration_79035988181304
// MI455X (gfx1250) — compile-verified
//
#include <hip/hip_runtime.h>
#include <hip/hip_bf16.h>
#include <math.h>

typedef __attribute__((ext_vector_type(2))) float v2f;
typedef __attribute__((ext_vector_type(8))) float v8f;
typedef __attribute__((ext_vector_type(4))) int   v4i;

#define NMAT 1024
#define BATCH 8
#define NNE ((size_t)NMAT * NMAT)   // 2^20

__device__ __forceinline__ v8f wmma_f32(v2f a, v2f b, v8f c) {
  // V_WMMA_F32_16X16X4_F32 : D = A(16x4) * B(4x16) + C(16x16), all fp32
  return __builtin_amdgcn_wmma_f32_16x16x4_f32(
      false, a, false, b, (short)0, c, false, false);
}

// ---- CDNA5 async global->LDS copy (ASYNCcnt-tracked), with safe fallback ----
#if __has_builtin(__builtin_amdgcn_global_load_async_to_lds_b128)
#define HAVE_ASYNC_LDS 1
__device__ __forceinline__ void async_copy16(const float* g, float* l) {
  __builtin_amdgcn_global_load_async_to_lds_b128(
      (__attribute__((address_space(1))) v4i*)(g),
      (__attribute__((address_space(3))) v4i*)(l), 0, 0);
}
#else
#define HAVE_ASYNC_LDS 0
__device__ __forceinline__ void async_copy16(const float* g, float* l) {
  *(float4*)l = *(const float4*)g;
}
#endif

__device__ __forceinline__ void wait_async0() {
#if HAVE_ASYNC_LDS
#if __has_builtin(__builtin_amdgcn_s_wait_asynccnt)
  __builtin_amdgcn_s_wait_asynccnt(0);
#else
  asm volatile("s_wait_asynccnt 0" ::: "memory");
#endif
#endif
}

// ---------------------------------------------------------------------------
// Pooling: out[b,i,j] = 0.5*(U[i,j]*G[b,i,j] + U[j,i]*G[b,j,i])
// ---------------------------------------------------------------------------
__global__ void pool_kernel(const float* __restrict__ G, const float* __restrict__ U,
                            float* __restrict__ out) {
  size_t idx = (size_t)blockIdx.x * blockDim.x + threadIdx.x;
  if (idx >= (size_t)BATCH * NNE) return;
  int c = (int)(idx & (NMAT - 1));
  int r = (int)((idx >> 10) & (NMAT - 1));
  size_t bbase = idx & ~(NNE - 1);
  float a  = U[(size_t)r * NMAT + c] * G[idx];
  float at = U[(size_t)c * NMAT + r] * G[bbase + (size_t)c * NMAT + r];
  out[idx] = 0.5f * (a + at);
}

// ---------------------------------------------------------------------------
// Row sums: deg[b*N + r] = sum_j Gp[b,r,j]   (one block per row)
// ---------------------------------------------------------------------------
__global__ void rowsum_kernel(const float* __restrict__ Gp, float* __restrict__ deg) {
  const float* row = Gp + (size_t)blockIdx.x * NMAT;
  int t = threadIdx.x;
  float s = 0.f;
  for (int j = t; j < NMAT; j += 256) s += row[j];
  __shared__ float red[256];
  red[t] = s; __syncthreads();
  for (int st = 128; st > 0; st >>= 1) {
    if (t < st) red[t] += red[t + st];
    __syncthreads();
  }
  if (t == 0) deg[blockIdx.x] = red[0];
}

// ---------------------------------------------------------------------------
// L[b,i,j] = (i==j ? deg : 0) - Gp[b,i,j]    (symmetric since Gp symmetric)
// ---------------------------------------------------------------------------
__global__ void buildL_kernel(const float* __restrict__ Gp, const float* __restrict__ deg,
                              float* __restrict__ L) {
  size_t idx = (size_t)blockIdx.x * blockDim.x + threadIdx.x;
  if (idx >= (size_t)BATCH * NNE) return;
  int c = (int)(idx & (NMAT - 1));
  int r = (int)((idx >> 10) & (NMAT - 1));
  float d = (r == c) ? deg[idx >> 10] : 0.f;
  L[idx] = d - Gp[idx];
}

// ---------------------------------------------------------------------------
// M = Gp + 1e-3*I   (LU input)
// ---------------------------------------------------------------------------
__global__ void copyeps_kernel(const float* __restrict__ Gp, float* __restrict__ M) {
  size_t idx = (size_t)blockIdx.x * blockDim.x + threadIdx.x;
  if (idx >= (size_t)BATCH * NNE) return;
  int c = (int)(idx & (NMAT - 1));
  int r = (int)((idx >> 10) & (NMAT - 1));
  M[idx] = Gp[idx] + ((r == c) ? 1e-3f : 0.f);
}

// ---------------------------------------------------------------------------
// BatchNorm1d over (B,N) per channel (256 channels), training stats
// ---------------------------------------------------------------------------
__global__ void bn_kernel(const float* __restrict__ R2, const float* __restrict__ g,
                          const float* __restrict__ be, float* __restrict__ out) {
  int c = blockIdx.x;
  int t = threadIdx.x;
  __shared__ float rs[256], rq[256];
  float s = 0.f, q = 0.f;
  const int M = BATCH * NMAT;  // 8192
  for (int i = t; i < M; i += 256) {
    float v = R2[(size_t)i * 256 + c];
    s += v; q += v * v;
  }
  rs[t] = s; rq[t] = q; __syncthreads();
  for (int st = 128; st > 0; st >>= 1) {
    if (t < st) { rs[t] += rs[t + st]; rq[t] += rq[t + st]; }
    __syncthreads();
  }
  float mu = rs[0] / (float)M;
  float var = rq[0] / (float)M - mu * mu;
  float sc = g[c] * rsqrtf(var + 1e-5f);
  float sh = be[c];
  for (int i = t; i < M; i += 256) {
    size_t p = (size_t)i * 256 + c;
    out[p] = (R2[p] - mu) * sc + sh;
  }
}

// ---------------------------------------------------------------------------
// fp32 WMMA GEMM. Block tile 128x64, K blocked by 64, double-buffered LDS
// filled with GLOBAL_LOAD_ASYNC_TO_LDS_B128. 8 waves/block; each wave owns a
// 32x32 strip = four 16x16 v8f accumulators (1 ds_load_b64 per WMMA).
// MODE 0: C = 2*(A@A) + A + 1  (cheb; A symmetric -> B tile is a row-major
//         tile of A, so both operands use straight async copies)
// MODE 1: C = A@Bm             (x2 = Lcheb @ R2n; B staged via transpose
//         scatter, A async)
// ---------------------------------------------------------------------------
template <int MODE>
__global__ void __launch_bounds__(256)
wmma_gemm_kernel(const float* __restrict__ A,
                 const float* __restrict__ Bm,
                 float* __restrict__ C, int ncols) {
  const int K = NMAT, lda = NMAT;
  int b = blockIdx.z;
  int m0 = blockIdx.y * 128;
  int n0 = blockIdx.x * 64;
  const float* Ab = A + (size_t)b * NNE;
  const float* Bb = Bm + (size_t)b * (size_t)NMAT * ncols;
  float* Cb = C + (size_t)b * (size_t)NMAT * ncols;

  // pitch 68: rows 16B-aligned (272B) for b128 copies; bank shift 4/row
  __shared__ float As[2][128][68];   // [m][k]
  __shared__ float Bs[2][64][68];    // MODE0: [n][k] (= rows of A); MODE1: [n][k] scatter

  int tid  = threadIdx.x;            // 0..255
  int lane = tid & 31;
  int w    = tid >> 5;               // wave 0..7
  int wm   = w & 3;                  // 4 row-groups of 32
  int wn   = w >> 2;                 // 2 col-groups of 32
  int half = lane >> 4;              // fp32 WMMA lane layout: K=2k / K=2k+2
  int lr   = lane & 15;

  int lrow = tid >> 4;               // 0..15 (loader row for MODE1 B)
  int lcol = (tid & 15) * 4;         // 0..60

  v8f a00 = {}, a01 = {}, a10 = {}, a11 = {};

  auto stage = [&](int p, int k0) {
    // A tile: 128x64 = 2048 b128 chunks, 8 per thread
#pragma unroll
    for (int q = 0; q < 8; ++q) {
      int cidx = tid + 256 * q;
      int r = cidx >> 4, c4 = (cidx & 15) << 2;
      async_copy16(Ab + (size_t)(m0 + r) * lda + k0 + c4, &As[p][r][c4]);
    }
    if constexpr (MODE == 0) {
      // B tile via symmetry: Bs[n][k] = A[n0+n][k0+k]  (row-major async)
#pragma unroll
      for (int q = 0; q < 4; ++q) {
        int cidx = tid + 256 * q;
        int r = cidx >> 4, c4 = (cidx & 15) << 2;
        async_copy16(Ab + (size_t)(n0 + r) * lda + k0 + c4, &Bs[p][r][c4]);
      }
    } else {
      // B tile: load rows of Bm, scatter transposed into Bs[n][k]
#pragma unroll
      for (int q = 0; q < 4; ++q) {
        int kr = lrow + 16 * q;
        float4 vb = *(const float4*)(Bb + (size_t)(k0 + kr) * ncols + n0 + lcol);
        Bs[p][lcol + 0][kr] = vb.x;
        Bs[p][lcol + 1][kr] = vb.y;
        Bs[p][lcol + 2][kr] = vb.z;
        Bs[p][lcol + 3][kr] = vb.w;
      }
    }
  };

  stage(0, 0);
  wait_async0();
  __syncthreads();

  int p = 0;
  for (int k0 = 0; k0 < K; k0 += 64) {
    if (k0 + 64 < K) stage(p ^ 1, k0 + 64);   // overlap next-tile loads

#pragma unroll
    for (int kk = 0; kk < 64; kk += 4) {
      int ka = kk + 2 * half;
      v2f fa0 = *(const v2f*)(&As[p][wm * 32 + lr][ka]);
      v2f fa1 = *(const v2f*)(&As[p][wm * 32 + 16 + lr][ka]);
      v2f fb0 = *(const v2f*)(&Bs[p][wn * 32 + lr][ka]);
      v2f fb1 = *(const v2f*)(&Bs[p][wn * 32 + 16 + lr][ka]);
      a00 = wmma_f32(fa0, fb0, a00);
      a01 = wmma_f32(fa0, fb1, a01);
      a10 = wmma_f32(fa1, fb0, a10);
      a11 = wmma_f32(fa1, fb1, a11);
    }
    wait_async0();
    __syncthreads();
    p ^= 1;
  }

  // D layout: VGPR r -> row (r + 8*half), col = lr within each 16x16 tile
  int gn0 = n0 + wn * 32 + lr;
  int gn1 = gn0 + 16;
#pragma unroll
  for (int r = 0; r < 8; ++r) {
    int mlo = r + (half ? 8 : 0);
    int gm0 = m0 + wm * 32 + mlo;
    int gm1 = gm0 + 16;
    if constexpr (MODE == 0) {
      Cb[(size_t)gm0 * ncols + gn0] = 2.f * a00[r] + Ab[(size_t)gm0 * lda + gn0] + 1.f;
      Cb[(size_t)gm0 * ncols + gn1] = 2.f * a01[r] + Ab[(size_t)gm0 * lda + gn1] + 1.f;
      Cb[(size_t)gm1 * ncols + gn0] = 2.f * a10[r] + Ab[(size_t)gm1 * lda + gn0] + 1.f;
      Cb[(size_t)gm1 * ncols + gn1] = 2.f * a11[r] + Ab[(size_t)gm1 * lda + gn1] + 1.f;
    } else {
      Cb[(size_t)gm0 * ncols + gn0] = a00[r];
      Cb[(size_t)gm0 * ncols + gn1] = a01[r];
      Cb[(size_t)gm1 * ncols + gn0] = a10[r];
      Cb[(size_t)gm1 * ncols + gn1] = a11[r];
    }
  }
}

// ---------------------------------------------------------------------------
// In-place LU (no pivoting) per 1024x1024 matrix; out[blk] = sum log|U_kk|.
// Column-parallel trailing update: coalesced global accesses; pivot row and
// factor column staged in LDS.
// ---------------------------------------------------------------------------
__global__ void __launch_bounds__(1024)
lu_logdet_kernel(float* __restrict__ M, float* __restrict__ out) {
  const int n = NMAT;
  float* A = M + (size_t)blockIdx.x * NNE;
  __shared__ float prow[NMAT];
  __shared__ float fcol[NMAT];
  int t = threadIdx.x;

  for (int k = 0; k < n - 1; ++k) {
    prow[t] = A[(size_t)k * n + t];
    __syncthreads();
    float pinv = 1.0f / prow[k];
    fcol[t] = (t > k) ? A[(size_t)t * n + k] * pinv : 0.f;
    __syncthreads();
    if (t > k) {
      float pj = prow[t];
      for (int i = k + 1; i < n; ++i)
        A[(size_t)i * n + t] -= fcol[i] * pj;  // coalesced across threads
    }
    __syncthreads();
  }
  prow[t] = logf(fabsf(A[(size_t)t * n + t]));
  __syncthreads();
  for (int st = 512; st > 0; st >>= 1) {
    if (t < st) prow[t] += prow[t + st];
    __syncthreads();
  }
  if (t == 0) out[blockIdx.x] = prow[0];
}

// ---------------------------------------------------------------------------
// x1pre[b,i,:] = Lcheb1[b,i,:] @ R1[b,:,0:4]   (one block per row)
// ---------------------------------------------------------------------------
__global__ void x1pre_kernel(const float* __restrict__ L, const float* __restrict__ R1,
                             float* __restrict__ out) {
  int row = blockIdx.x;             // b*1024 + i
  int bb = row >> 10;
  int t = threadIdx.x;
  const float* Lr = L + (size_t)row * NMAT;
  const float* Rb = R1 + (size_t)bb * NMAT * 4;
  float acc0 = 0, acc1 = 0, acc2 = 0, acc3 = 0;
  for (int k = t; k < NMAT; k += 256) {
    float lv = Lr[k];
    acc0 += lv * Rb[k * 4 + 0];
    acc1 += lv * Rb[k * 4 + 1];
    acc2 += lv * Rb[k * 4 + 2];
    acc3 += lv * Rb[k * 4 + 3];
  }
  __shared__ float red[4][256];
  red[0][t] = acc0; red[1][t] = acc1; red[2][t] = acc2; red[3][t] = acc3;
  __syncthreads();
  for (int st = 128; st > 0; st >>= 1) {
    if (t < st)
      for (int c2 = 0; c2 < 4; ++c2) red[c2][t] += red[c2][t + st];
    __syncthreads();
  }
  if (t < 4) out[(size_t)row * 4 + t] = red[t][0];
}

// ---------------------------------------------------------------------------
// Conv1d(4->256, K=5, pad=2) over length N + bias
// ---------------------------------------------------------------------------
__global__ void conv_kernel(const float* __restrict__ xp, const float* __restrict__ w,
                            const float* __restrict__ bias, float* __restrict__ out) {
  size_t idx = (size_t)blockIdx.x * blockDim.x + threadIdx.x;
  if (idx >= (size_t)BATCH * NMAT * 256) return;
  int co = (int)(idx & 255);
  int n  = (int)((idx >> 8) & (NMAT - 1));
  int bb = (int)(idx >> 18);
  float s = bias[co];
  const float* wc = w + (size_t)co * 20;  // [ci][t] = 4x5
#pragma unroll
  for (int tt = 0; tt < 5; ++tt) {
    int nn = n + tt - 2;
    if (nn >= 0 && nn < NMAT) {
      const float* xr = xp + ((size_t)bb * NMAT + nn) * 4;
#pragma unroll
      for (int ci = 0; ci < 4; ++ci) s += xr[ci] * wc[ci * 5 + tt];
    }
  }
  out[idx] = s;
}

// ---------------------------------------------------------------------------
// Wasserstein: per (b,c) column: bitonic-sort both 1024-length columns of
// x1/x2 in LDS, partial[col] = sum |sorted1 - sorted2|
// ---------------------------------------------------------------------------
__global__ void __launch_bounds__(512)
wass_kernel(const float* __restrict__ x1, const float* __restrict__ x2,
            float* __restrict__ partial) {
  __shared__ float s1[1024];
  __shared__ float s2[1024];
  int col = blockIdx.x;            // b*256 + c
  int bb = col >> 8;
  int c  = col & 255;
  const float* p1 = x1 + (size_t)bb * NMAT * 256 + c;
  const float* p2 = x2 + (size_t)bb * NMAT * 256 + c;
  int t = threadIdx.x;             // 0..511
  s1[t]       = p1[(size_t)t * 256];
  s1[t + 512] = p1[(size_t)(t + 512) * 256];
  s2[t]       = p2[(size_t)t * 256];
  s2[t + 512] = p2[(size_t)(t + 512) * 256];

  for (unsigned k = 2; k <= 1024; k <<= 1) {
    for (unsigned j = k >> 1; j > 0; j >>= 1) {
      __syncthreads();
      unsigned i = 2u * t - (t & (j - 1));
      unsigned ij = i + j;
      bool up = ((i & k) == 0);
      float a1 = s1[i], b1 = s1[ij];
      if ((a1 > b1) == up) { s1[i] = b1; s1[ij] = a1; }
      float a2 = s2[i], b2 = s2[ij];
      if ((a2 > b2) == up) { s2[i] = b2; s2[ij] = a2; }
    }
  }
  __syncthreads();
  float d = fabsf(s1[t] - s2[t]) + fabsf(s1[t + 512] - s2[t + 512]);
  __syncthreads();
  s1[t] = d;
  __syncthreads();
  for (int st = 256; st > 0; st >>= 1) {
    if (t < st) s1[t] += s1[t + st];
    __syncthreads();
  }
  if (t == 0) partial[col] = s1[0];
}

// ---------------------------------------------------------------------------
// Final losses (deterministic single-block reduction)
// ---------------------------------------------------------------------------
__global__ void finalize_kernel(const float* __restrict__ ld,
                                const float* __restrict__ wpart,
                                float* __restrict__ loss_out) {
  __shared__ float red[256];
  int t = threadIdx.x;
  float s = 0.f;
  for (int i = t; i < 2048; i += 256) s += wpart[i];
  red[t] = s; __syncthreads();
  for (int st = 128; st > 0; st >>= 1) {
    if (t < st) red[t] += red[t + st];
    __syncthreads();
  }
  if (t == 0) {
    float l1 = 0.f, l2 = 0.f;
    for (int i = 0; i < 8; ++i) { l1 += ld[i]; l2 += ld[8 + i]; }
    loss_out[0] = -l1 / 8.f;
    loss_out[1] = -l2 / 8.f;
    loss_out[2] = red[0] / (float)((size_t)BATCH * NMAT * 256);
  }
}

// ---------------------------------------------------------------------------
extern "C" void kernel_launch(void* const* d_in, const int* in_sizes, int n_in,
                              void* d_out, int out_size, void* d_ws, size_t ws_size,
                              hipStream_t stream) {
  (void)in_sizes; (void)n_in; (void)out_size; (void)ws_size;
  const float* G1  = (const float*)d_in[0];
  const float* G2  = (const float*)d_in[1];
  const float* R1  = (const float*)d_in[2];
  const float* R2  = (const float*)d_in[3];
  const float* U1  = (const float*)d_in[7];
  const float* U2  = (const float*)d_in[8];
  const float* cw  = (const float*)d_in[9];
  const float* cb  = (const float*)d_in[10];
  const float* bg  = (const float*)d_in[11];
  const float* bbt = (const float*)d_in[12];

  float* out = (float*)d_out;
  float* G1p = out;
  float* G2p = out + 8 * NNE;
  float* x1  = out + 16 * NNE;
  float* x2  = out + 16 * NNE + 2 * NNE;
  float* losses = out + 20 * NNE;

  float* w = (float*)d_ws;
  float* L1    = w;                       // 8*NNE   (later reused as LU scratch 1)
  float* L2    = w + 8 * NNE;             // 8*NNE   (later reused as LU scratch 2)
  float* C1    = w + 16 * NNE;            // 8*NNE
  float* C2    = w + 24 * NNE;            // 8*NNE
  float* deg1  = w + 32 * NNE;            // 8192
  float* deg2  = deg1 + 8192;             // 8192
  float* R2n   = deg2 + 8192;             // 2*NNE
  float* x1pre = R2n + 2 * NNE;           // 32768
  float* ld    = x1pre + 32768;           // 16
  float* wpart = ld + 16;                 // 2048

  const int ew = (int)((BATCH * NNE + 255) / 256);  // elementwise blocks: 32768

  // 1) pooling -> outputs G1p / G2p
  pool_kernel<<<ew, 256, 0, stream>>>(G1, U1, G1p);
  pool_kernel<<<ew, 256, 0, stream>>>(G2, U2, G2p);

  // 2) degrees + Laplacians
  rowsum_kernel<<<BATCH * NMAT, 256, 0, stream>>>(G1p, deg1);
  rowsum_kernel<<<BATCH * NMAT, 256, 0, stream>>>(G2p, deg2);
  buildL_kernel<<<ew, 256, 0, stream>>>(G1p, deg1, L1);
  buildL_kernel<<<ew, 256, 0, stream>>>(G2p, deg2, L2);

  // 3) batchnorm R2 -> R2n
  bn_kernel<<<256, 256, 0, stream>>>(R2, bg, bbt, R2n);

  // 4) chebyshev: C = 2*L@L + L + 1   (fp32 WMMA, async double-buffered)
  wmma_gemm_kernel<0><<<dim3(16, 8, BATCH), 256, 0, stream>>>(L1, L1, C1, NMAT);
  wmma_gemm_kernel<0><<<dim3(16, 8, BATCH), 256, 0, stream>>>(L2, L2, C2, NMAT);

  // 5) slogdet via LU (L1/L2 buffers now free -> reuse as LU scratch)
  copyeps_kernel<<<ew, 256, 0, stream>>>(G1p, L1);
  copyeps_kernel<<<ew, 256, 0, stream>>>(G2p, L2);
  lu_logdet_kernel<<<BATCH, 1024, 0, stream>>>(L1, ld);
  lu_logdet_kernel<<<BATCH, 1024, 0, stream>>>(L2, ld + 8);

  // 6) x1 = conv1d(L1cheb @ R1)
  x1pre_kernel<<<BATCH * NMAT, 256, 0, stream>>>(C1, R1, x1pre);
  conv_kernel<<<(int)((2 * NNE + 255) / 256), 256, 0, stream>>>(x1pre, cw, cb, x1);

  // 7) x2 = L2cheb @ R2n   (fp32 WMMA, ncols = 256)
  wmma_gemm_kernel<1><<<dim3(4, 8, BATCH), 256, 0, stream>>>(C2, R2n, x2, 256);

  // 8) Wasserstein partials + final losses (deterministic)
  wass_kernel<<<BATCH * 256, 512, 0, stream>>>(x1, x2, wpart);
  finalize_kernel<<<1, 256, 0, stream>>>(ld, wpart, losses);
}